// DeformConv2D_5995774345871
// MI455X (gfx1250) — compile-verified
//
#include <hip/hip_runtime.h>

#define BATCH 4
#define CH 64
#define HH 128
#define WW 128
#define HW (HH * WW)        // 16384
#define NOFF 9              // 3x3 sampling points
#define KDIM (CH * NOFF)    // 576 contraction length
#define OCH 64
#define TILE_M 16
#define ASTRIDE 577         // 576 + 1 pad -> avoid 64-bank conflicts

typedef float v2f __attribute__((ext_vector_type(2)));
typedef float v8f __attribute__((ext_vector_type(8)));

// ---------------- Kernel A: offset conv (3x3, pad 1, 64 -> 18 ch) ----------
__global__ void offset_conv_kernel(const float* __restrict__ x,
                                   const float* __restrict__ Woff,
                                   const float* __restrict__ boff,
                                   float* __restrict__ off) {
  int gid = blockIdx.x * blockDim.x + threadIdx.x;
  if (gid >= BATCH * HW) return;
  int b = gid >> 14;
  int pix = gid & (HW - 1);
  int i = pix >> 7, j = pix & 127;
  float acc[18];
#pragma unroll
  for (int ch = 0; ch < 18; ++ch) acc[ch] = boff[ch];
  const float* xb = x + (size_t)b * CH * HW;
  for (int c = 0; c < CH; ++c) {
#pragma unroll
    for (int k1 = 0; k1 < 3; ++k1) {
      int y = i + k1 - 1;
#pragma unroll
      for (int k2 = 0; k2 < 3; ++k2) {
        int xx = j + k2 - 1;
        float xv = 0.f;
        if (y >= 0 && y < HH && xx >= 0 && xx < WW)
          xv = xb[c * HW + (y << 7) + xx];
        int wbase = c * 9 + k1 * 3 + k2;
#pragma unroll
        for (int ch = 0; ch < 18; ++ch)
          acc[ch] = fmaf(xv, Woff[ch * KDIM + wbase], acc[ch]);
      }
    }
  }
#pragma unroll
  for (int ch = 0; ch < 18; ++ch)
    off[((size_t)b * 18 + ch) * HW + pix] = acc[ch];
}

// -------- Kernel T: W_conv[oc][k] -> Wtp[k/2][oc][2] (K-pair interleave) ---
// Lets each lane in the GEMM fetch its (K, K+1) B-operand pair with one
// 8-byte global_load_b64 instead of two strided b32 loads.
__global__ void wt_kernel(const float* __restrict__ Wc, float* __restrict__ Wtp) {
  int gid = blockIdx.x * blockDim.x + threadIdx.x;
  if (gid >= OCH * KDIM) return;
  int oc = gid / KDIM;
  int k  = gid - oc * KDIM;
  Wtp[(k >> 1) * (2 * OCH) + oc * 2 + (k & 1)] = Wc[gid];
}

// ---------------- Kernel B: fused bilinear gather + f32 WMMA GEMM ----------
__global__ __launch_bounds__(128)
void deform_gemm_kernel(const float* __restrict__ x,
                        const float* __restrict__ off,
                        const float* __restrict__ Wtp,
                        float* __restrict__ out) {
  __shared__ float As[TILE_M * ASTRIDE];   // 36,928 B; reused for D staging
  int tid = threadIdx.x;
  int blk = blockIdx.x;                    // 0..4095
  int b = blk >> 10;                       // 1024 tiles of 16 pixels per batch
  int tile = blk & 1023;
  int pixbase = tile * TILE_M;             // 16 divides 128 -> one image row
  int i0 = pixbase >> 7;
  int j0 = pixbase & 127;
  const float* xb = x + (size_t)b * CH * HW;
  const float* offb = off + (size_t)b * 18 * HW;

  // ---- Phase 1: build A tile (16 pixels x 576 K) in LDS --------------------
  for (int pair = tid; pair < TILE_M * NOFF; pair += 128) {
    int m = pair / NOFF;
    int n = pair - m * NOFF;
    int j = j0 + m;
    int pix = pixbase + m;
    float offx = offb[(2 * n) * HW + pix];       // row-offset (channel 2n)
    float offy = offb[(2 * n + 1) * HW + pix];   // col-offset (channel 2n+1)
    // p = p0 + p_n + offset, padded coords (Hp=Wp=130)
    float px = (float)(i0 + (n / 3)) + offx;     // (i+1) + (n/3 - 1) + off
    float py = (float)(j  + (n % 3)) + offy;
    float flx = floorf(px), fly = floorf(py);
    float q0r = fminf(fmaxf(flx,       0.f), 129.f);
    float q0c = fminf(fmaxf(fly,       0.f), 129.f);
    float q1r = fminf(fmaxf(flx + 1.f, 0.f), 129.f);
    float q1c = fminf(fmaxf(fly + 1.f, 0.f), 129.f);
    // reference mask trick: snap p to floor(p) outside [PAD, Hp-1-PAD], clip
    if (px < 1.f || px > 128.f) px = flx;
    if (py < 1.f || py > 128.f) py = fly;
    px = fminf(fmaxf(px, 0.f), 129.f);
    py = fminf(fmaxf(py, 0.f), 129.f);
    float g_lt = (1.f + (q0r - px)) * (1.f + (q0c - py));
    float g_rb = (1.f - (q1r - px)) * (1.f - (q1c - py));
    float g_lb = (1.f + (q0r - px)) * (1.f - (q1c - py));
    float g_rt = (1.f - (q1r - px)) * (1.f + (q0c - py));
    int r0 = (int)q0r - 1, c0 = (int)q0c - 1;    // padded -> unpadded coords
    int r1 = (int)q1r - 1, c1 = (int)q1c - 1;
    bool okr0 = (r0 >= 0) & (r0 < HH), okc0 = (c0 >= 0) & (c0 < WW);
    bool okr1 = (r1 >= 0) & (r1 < HH), okc1 = (c1 >= 0) & (c1 < WW);
    int l_lt = (r0 << 7) + c0, l_rb = (r1 << 7) + c1;
    int l_lb = (r0 << 7) + c1, l_rt = (r1 << 7) + c0;
    bool ok_lt = okr0 & okc0, ok_rb = okr1 & okc1;
    bool ok_lb = okr0 & okc1, ok_rt = okr1 & okc0;
    float* arow = &As[m * ASTRIDE + n];
    for (int c = 0; c < CH; ++c) {
      const float* xc = xb + c * HW;
      float v = 0.f;
      if (ok_lt) v = fmaf(g_lt, xc[l_lt], v);
      if (ok_rb) v = fmaf(g_rb, xc[l_rb], v);
      if (ok_lb) v = fmaf(g_lb, xc[l_lb], v);
      if (ok_rt) v = fmaf(g_rt, xc[l_rt], v);
      arow[c * NOFF] = v;                        // K index = c*9 + n
    }
  }
  __syncthreads();

  // ---- Phase 2: D(16x16) = A(16x576) x Wt(576x16) via v_wmma_f32_16x16x4 --
  int wave = tid >> 5;
  int lane = tid & 31;
  int half = lane >> 4;      // ISA 32-bit A 16x4: VGPR0 <- K={0,2}, VGPR1 <- K={1,3}
  int l16  = lane & 15;      // M (for A) / N (for B) = lane % 16
  int ocb  = wave * 16;
  const float* arow = &As[l16 * ASTRIDE];
  const v2f* brow = ((const v2f*)Wtp) + (ocb + l16);  // pair-interleaved B
  v8f acc = {0.f, 0.f, 0.f, 0.f, 0.f, 0.f, 0.f, 0.f};
#pragma unroll 4
  for (int k = 0; k < KDIM; k += 4) {
    int ka = k + 2 * half;
    v2f a, bm;
    a.x = arow[ka];
    a.y = arow[ka + 1];
    bm  = brow[(size_t)(ka >> 1) * OCH];   // one global_load_b64: K=ka, ka+1
    acc = __builtin_amdgcn_wmma_f32_16x16x4_f32(
        /*neg_a=*/false, a, /*neg_b=*/false, bm,
        /*c_mod=*/(short)0, acc, /*reuse_a=*/false, /*reuse_b=*/false);
  }
  __syncthreads();           // all waves done reading As before reuse

  // ---- Phase 3: stage D through LDS, coalesced store ----------------------
  float* Ds = As;            // 16 x 65 floats, padded stride
#pragma unroll
  for (int v = 0; v < 8; ++v) {
    int m = v + 8 * half;    // C/D layout: VGPR v -> M = v (lanes 0-15) / v+8
    Ds[m * 65 + ocb + l16] = acc[v];
  }
  __syncthreads();
  float* outb = out + (size_t)b * OCH * HW;
  for (int e = tid; e < TILE_M * OCH; e += 128) {
    int m  = e & 15;
    int oc = e >> 4;
    outb[oc * HW + pixbase + m] = Ds[m * 65 + oc];
  }
}

extern "C" void kernel_launch(void* const* d_in, const int* in_sizes, int n_in,
                              void* d_out, int out_size, void* d_ws, size_t ws_size,
                              hipStream_t stream) {
  const float* x    = (const float*)d_in[0];  // (4,64,128,128)
  const float* Woff = (const float*)d_in[1];  // (18,64,3,3)
  const float* boff = (const float*)d_in[2];  // (18,)
  const float* Wc   = (const float*)d_in[3];  // (64,64,3,3)
  float* out = (float*)d_out;                 // (4,64,128,128)

  float* off = (float*)d_ws;                                   // 4*18*16384 f32
  float* Wtp = (float*)((char*)d_ws +
                        (size_t)BATCH * 18 * HW * sizeof(float)); // 576*64 f32

  offset_conv_kernel<<<(BATCH * HW) / 256, 256, 0, stream>>>(x, Woff, boff, off);
  wt_kernel<<<(OCH * KDIM + 255) / 256, 256, 0, stream>>>(Wc, Wtp);
  deform_gemm_kernel<<<BATCH * (HW / TILE_M), 128, 0, stream>>>(x, off, Wtp, out);
}